// BlockContrastiveLoss_21835613733421
// MI455X (gfx1250) — compile-verified
//
#include <hip/hip_runtime.h>
#include <hip/hip_bf16.h>
#include <stdint.h>

// BlockContrastiveLoss for MI455X (gfx1250).
// semantic_state: f32[32,8192,64] -> 262144 rows of 16 blocks x 4 floats.
// token_ids: harness passes integers as const int* (per harness contract).
// d_out: 1 float (the scalar loss).

#define NROWS        (32 * 8192)          // 262144 token rows
#define ROWF         64                   // floats per row
#define VOCAB_       1024
#define BINSZ        (VOCAB_ * ROWF)      // 65536 floats of segment sums
#define PARTSZ       (BINSZ + VOCAB_)     // + 1024 counts = 66560 floats
#define THREADS      256
#define WAVES        8
#define UNR          4                    // b128 copies per buffer per wave (8 rows)
#define ROWS_PER_IT  64                   // rows consumed per WG iteration
#define ITER_TOTAL   (NROWS / ROWS_PER_IT) // 4096
#define EPS_         1e-12f

// ---------------- CDNA5 async global->LDS path ----------------
#if defined(__has_builtin)
#if __has_builtin(__builtin_amdgcn_global_load_async_to_lds_b128)
#define HAVE_ASYNC 1
#endif
#if __has_builtin(__builtin_amdgcn_wmma_f32_16x16x4_f32)
#define HAVE_WMMA 1
#endif
#endif
#ifndef HAVE_ASYNC
#define HAVE_ASYNC 0
#endif
#ifndef HAVE_WMMA
#define HAVE_WMMA 0
#endif

typedef int   v4i_ __attribute__((ext_vector_type(4)));
typedef float v2f_ __attribute__((ext_vector_type(2)));
typedef float v8f_ __attribute__((ext_vector_type(8)));

#if HAVE_ASYNC
__device__ __forceinline__ void asyncCopy16(const float* g, float* l) {
  // (global v4i*, lds v4i*, imm offset, imm cpol); tracked by ASYNCcnt
  __builtin_amdgcn_global_load_async_to_lds_b128(
      (__attribute__((address_space(1))) v4i_*)g,
      (__attribute__((address_space(3))) v4i_*)l,
      0, 0);
}
template <int N>
__device__ __forceinline__ void waitAsync() {
#if __has_builtin(__builtin_amdgcn_s_wait_asynccnt)
  __builtin_amdgcn_s_wait_asynccnt(N);
#else
  asm volatile("s_wait_asynccnt %0" ::"i"(N) : "memory");
#endif
}
#endif

// ---------------- Phase 1: stream + LDS-privatized segment sum ----------------
__global__ __launch_bounds__(THREADS) void bcl_phase1(
    const float* __restrict__ sem, const int* __restrict__ tok,
    float* __restrict__ ws, int G, int usePartials) {
  __shared__ alignas(16) float bins[BINSZ];   // 256 KB
  __shared__ alignas(16) float cnt[VOCAB_];   //   4 KB
#if HAVE_ASYNC
  __shared__ alignas(16) float stage[WAVES * 2 * UNR * 128];  // 32 KB
#endif
  for (int i = threadIdx.x; i < BINSZ; i += THREADS) bins[i] = 0.f;
  for (int i = threadIdx.x; i < VOCAB_; i += THREADS) cnt[i] = 0.f;
  __syncthreads();

  const int lane = threadIdx.x & 31;
  const int wave = threadIdx.x >> 5;
  const int blk  = lane & 15;   // which 4-float block of the row
  const int rsub = lane >> 4;   // which of the 2 rows this wave-load covers

  auto process = [&](int r, float x, float y, float z, float w) {
    const int t = tok[r];
    const float ss  = x * x + y * y + z * z + w * w;
    const float inv = 1.0f / fmaxf(sqrtf(ss), EPS_);  // matches F.normalize eps
    float* d = &bins[t * ROWF + blk * 4];
    atomicAdd(d + 0, x * inv);   // ds_add_f32
    atomicAdd(d + 1, y * inv);
    atomicAdd(d + 2, z * inv);
    atomicAdd(d + 3, w * inv);
    if (blk == 0) atomicAdd(&cnt[t], 1.0f);
  };

#if HAVE_ASYNC
  float* wst = &stage[wave * (2 * UNR * 128)];
  auto issue = [&](int it, int b) {
    const float* g = sem + (size_t)it * (ROWS_PER_IT * ROWF) +
                     wave * (8 * ROWF) + rsub * ROWF + blk * 4;
    float* l = wst + b * (UNR * 128) + lane * 4;
#pragma unroll
    for (int u = 0; u < UNR; ++u) asyncCopy16(g + u * 128, l + u * 128);
  };
  auto consume = [&](int it, int b) {
    const int rbase = it * ROWS_PER_IT + wave * 8 + rsub;
    const float* l  = wst + b * (UNR * 128) + lane * 4;
#pragma unroll
    for (int u = 0; u < UNR; ++u) {
      const float4 v = *(const float4*)(l + u * 128);  // ds_load_b128
      process(rbase + u * 2, v.x, v.y, v.z, v.w);
    }
  };
  int it = blockIdx.x;
  if (it < ITER_TOTAL) {
    issue(it, 0);
    int b = 0;
    while (true) {
      const int itn = it + G;
      const bool more = itn < ITER_TOTAL;
      if (more) { issue(itn, b ^ 1); waitAsync<UNR>(); }  // keep next buf in flight
      else      { waitAsync<0>(); }
      consume(it, b);
      if (!more) break;
      it = itn;
      b ^= 1;
    }
  }
#else
  for (int it = blockIdx.x; it < ITER_TOTAL; it += G) {
    const float* g = sem + (size_t)it * (ROWS_PER_IT * ROWF) +
                     wave * (8 * ROWF) + rsub * ROWF + blk * 4;
    const int rbase = it * ROWS_PER_IT + wave * 8 + rsub;
#pragma unroll
    for (int u = 0; u < UNR; ++u) {
      const float4 v = *(const float4*)(g + u * 128);  // global_load_b128
      process(rbase + u * 2, v.x, v.y, v.z, v.w);
    }
  }
#endif

  __syncthreads();
  if (usePartials) {
    float4* dst = (float4*)(ws + (size_t)PARTSZ * (1 + blockIdx.x));
    const float4* b4 = (const float4*)bins;
    for (int i = threadIdx.x; i < BINSZ / 4; i += THREADS) dst[i] = b4[i];
    const float4* c4 = (const float4*)cnt;
    float4* dc = dst + BINSZ / 4;
    for (int i = threadIdx.x; i < VOCAB_ / 4; i += THREADS) dc[i] = c4[i];
  } else {
    for (int i = threadIdx.x; i < BINSZ; i += THREADS) {
      const float v = bins[i];
      if (v != 0.f) atomicAdd(&ws[i], v);
    }
    for (int i = threadIdx.x; i < VOCAB_; i += THREADS) {
      const float v = cnt[i];
      if (v != 0.f) atomicAdd(&ws[BINSZ + i], v);
    }
  }
}

// ---------------- Phase 2: reduce partials (L2-resident) ----------------
__global__ __launch_bounds__(256) void bcl_reduce(float* __restrict__ ws, int G) {
  const int o = blockIdx.x * 256 + threadIdx.x;
  if (o >= PARTSZ) return;
  float s = 0.f;
  const float* p = ws + PARTSZ + o;
  for (int g = 0; g < G; ++g) s += p[(size_t)g * PARTSZ];
  ws[o] = s;
}

// ---------------- Phase 3: closed-form loss ----------------
// ss_v = sum_k seg[v][k]^2 is the diagonal of the 16x16 Gram matrix S*S^T of
// each 16-row tile: computed with v_wmma_f32_16x16x4_f32 accumulating over 16
// k-chunks. With B = A^T, the documented A (16x4) and B (4x16) VGPR layouts
// coincide, so one register pair feeds both operands.
__global__ __launch_bounds__(1024) void bcl_final(const float* __restrict__ ws,
                                                  float* __restrict__ out) {
  __shared__ float acc[2];
  const int tid = threadIdx.x;
  if (tid == 0) { acc[0] = 0.f; acc[1] = 0.f; }
  __syncthreads();
#if HAVE_WMMA
  const int lane = tid & 31;
  const int wave = tid >> 5;   // 32 waves, 2 tiles each -> 64 tiles = 1024 tokens
  float num = 0.f, pr = 0.f;
  for (int t = 0; t < 2; ++t) {
    const int tile = wave * 2 + t;
    const int row  = tile * 16 + (lane & 15);
    // lanes 0-15: K = kc*4 + {0,1};  lanes 16-31: K = kc*4 + {2,3}
    const float* base = ws + (size_t)row * ROWF + (lane >> 4) * 2;
    v8f_ c = {0.f, 0.f, 0.f, 0.f, 0.f, 0.f, 0.f, 0.f};
#pragma unroll
    for (int kc = 0; kc < 16; ++kc) {
      const v2f_ ab = *(const v2f_*)(base + kc * 4);
      c = __builtin_amdgcn_wmma_f32_16x16x4_f32(false, ab, false, ab,
                                                (short)0, c, false, false);
    }
    // diag D[v][v]: v=0..7 -> lane v, vgpr v ; v=8..15 -> lane v+16, vgpr v-8
    int v = -1, idx = 0;
    if (lane < 8)        { v = tile * 16 + lane;        idx = lane; }
    else if (lane >= 24) { v = tile * 16 + (lane - 16); idx = lane - 24; }
    if (v >= 0) {
      float ss = 0.f;
#pragma unroll
      for (int j = 0; j < 8; ++j)
        if (idx == j) ss = c[j];
      const float cn = ws[BINSZ + v];
      if (cn >= 2.f) num += 0.03125f * (ss - 16.f * cn);
      pr += 0.5f * cn * (cn - 1.f);
    }
  }
  atomicAdd(&acc[0], num);
  atomicAdd(&acc[1], pr);
#else
  const int v = tid;  // one token id per thread
  const float cn = ws[BINSZ + v];
  const float4* row = (const float4*)(ws + v * ROWF);
  float ss = 0.f;
#pragma unroll
  for (int k = 0; k < 16; ++k) {
    const float4 q = row[k];
    ss += q.x * q.x + q.y * q.y + q.z * q.z + q.w * q.w;
  }
  const float num = (cn >= 2.f) ? 0.03125f * (ss - 16.f * cn) : 0.f;
  const float pr  = 0.5f * cn * (cn - 1.f);
  atomicAdd(&acc[0], num);
  atomicAdd(&acc[1], pr);
#endif
  __syncthreads();
  if (tid == 0) out[0] = (acc[1] > 0.f) ? acc[0] / fmaxf(acc[1], 1.f) : 0.f;
}

extern "C" void kernel_launch(void* const* d_in, const int* in_sizes, int n_in,
                              void* d_out, int out_size, void* d_ws, size_t ws_size,
                              hipStream_t stream) {
  const float* sem = (const float*)d_in[0];
  const int*   tok = (const int*)d_in[1];
  float* out = (float*)d_out;
  float* ws  = (float*)d_ws;

  const size_t partBytes = (size_t)PARTSZ * sizeof(float);
  int G, useP;
  if (ws_size >= 9 * partBytes) {
    size_t g = ws_size / partBytes - 1;  // [final][G partials]
    if (g > 256) g = 256;
    G = (int)g;
    useP = 1;
  } else {
    G = 64;
    useP = 0;
    (void)hipMemsetAsync(d_ws, 0, partBytes, stream);  // capture-safe zero for atomics
  }

  bcl_phase1<<<G, THREADS, 0, stream>>>(sem, tok, ws, G, useP);
  if (useP) bcl_reduce<<<(PARTSZ + 255) / 256, 256, 0, stream>>>(ws, G);
  bcl_final<<<1, 1024, 0, stream>>>(ws, out);
}